// PortillaSimoncelli_20203526160914
// MI455X (gfx1250) — compile-verified
//
#include <hip/hip_runtime.h>
#include <math.h>

// ---------------------------------------------------------------------------
// Portilla-Simoncelli texture statistics for MI455X (gfx1250, wave32).
// All 2-D (I)FFTs are executed as DFT matrix products on the matrix cores via
// V_WMMA_F32_16X16X4_F32 (fp32 WMMA). Spectra are kept in UNSHIFTED layout.
// Pass-1 GEMMs (shared DFT-matrix B operand) stage the B panel into LDS with
// the Tensor Data Mover (tensor_load_to_lds + s_wait_tensorcnt), shared by all
// waves of the workgroup; other shapes use the branch-free global-path GEMM.
// ---------------------------------------------------------------------------

#define PSB   32      // batch (images)
#define PSNS  4       // scales
#define PSNO  4       // orientations
#define PI_F  3.14159265358979323846f

typedef __attribute__((ext_vector_type(2))) float v2f;
typedef __attribute__((ext_vector_type(8))) float v8f;
typedef unsigned int u32x4 __attribute__((ext_vector_type(4)));
typedef int i32x8 __attribute__((ext_vector_type(8)));
typedef int i32x4 __attribute__((ext_vector_type(4)));

// ------------------------------ reductions ---------------------------------
__device__ inline float blockReduceSum(float v, float* sh) {
    int t = threadIdx.x;
    sh[t] = v; __syncthreads();
    for (int w = 128; w > 0; w >>= 1) {
        if (t < w) sh[t] += sh[t + w];
        __syncthreads();
    }
    float r = sh[0]; __syncthreads();
    return r;
}

// --------------------------- DFT matrix builder ----------------------------
__global__ void k_make_dft(float* Dr, float* Di, float* Ir, float* Ii, int n) {
    int idx = blockIdx.x * blockDim.x + threadIdx.x;
    if (idx >= n * n) return;
    int j = idx / n, k = idx % n;
    long long jk = ((long long)j * (long long)k) % n;
    double ang = -2.0 * 3.14159265358979323846 * (double)jk / (double)n;
    float c = (float)cos(ang), s = (float)sin(ang);
    Dr[idx] = c;            // forward: e^{-2 pi i jk/n}
    Di[idx] = s;
    Ir[idx] = c / (float)n; // inverse: conj / n
    Ii[idx] = -s / (float)n;
}

// ----------------------- radial filters (unshifted) ------------------------
__global__ void k_radial(float* out, int n, float shiftc, int isSin) {
    int idx = blockIdx.x * blockDim.x + threadIdx.x;
    if (idx >= n * n) return;
    int x = idx % n, y = idx / n;
    int xs = (x + n / 2) % n, ys = (y + n / 2) % n;
    float xv = (float)(xs - n / 2) * (2.0f / n);
    float yv = (float)(ys - n / 2) * (2.0f / n);
    float rad = sqrtf(xv * xv + yv * yv);
    if (xs == n / 2 && ys == n / 2) rad = 2.0f / n;   // center fix: rad[cy,cx-1]
    float u = log2f(rad) + shiftc;
    u = fminf(fmaxf(u, 0.0f), 1.0f);
    out[idx] = isSin ? sinf(0.5f * PI_F * u) : cosf(0.5f * PI_F * u);
}

// ------------- orientation band filters (him * amask, unshifted) -----------
__global__ void k_bands(float* fb, int n) {
    int idx = blockIdx.x * blockDim.x + threadIdx.x;
    if (idx >= n * n) return;
    int x = idx % n, y = idx / n;
    int xs = (x + n / 2) % n, ys = (y + n / 2) % n;
    float xv = (float)(xs - n / 2) * (2.0f / n);
    float yv = (float)(ys - n / 2) * (2.0f / n);
    float rad = sqrtf(xv * xv + yv * yv);
    if (xs == n / 2 && ys == n / 2) rad = 2.0f / n;
    float u = log2f(rad) + 2.0f;
    u = fminf(fmaxf(u, 0.0f), 1.0f);
    float him = sinf(0.5f * PI_F * u);
    float ang = atan2f(yv, xv);
    float amp = 2.0f * sqrtf(0.8f);   // 2*sqrt(const), const=0.8 for NO=4
    for (int b = 0; b < 4; ++b) {
        float ad = ang - PI_F * (float)b * 0.25f + PI_F;
        ad = ad - floorf(ad / (2.0f * PI_F)) * (2.0f * PI_F);
        ad -= PI_F;
        float c = cosf(ad);
        float m = (fabsf(ad) < 0.5f * PI_F) ? (amp * c * c * c) : 0.0f;
        fb[(long long)b * n * n + idx] = him * m;
    }
}

// ------------------- TDM: async 2-D panel load into LDS --------------------
// D# per CDNA5 ISA ch.8: group0 = {flags, lds_addr, global_addr, type=2},
// group1 = {data_size=4B, tensor_dim0/1, tile_dim0/1, tensor_dim0_stride}.
// clang-23 toolchain: 6-arg builtin (g0, g1, g2, g3, extra, cpol).
__device__ inline void tdm_load_panel(const float* gsrc, unsigned lds_byte_off,
                                      int rows, int cols, int ldb) {
    unsigned long long ga = (unsigned long long)(size_t)gsrc;
    u32x4 g0;
    g0[0] = 1u;                                        // count=1 (valid user D#)
    g0[1] = lds_byte_off;                              // lds_addr (bytes)
    g0[2] = (unsigned)(ga & 0xffffffffu);              // global_addr[31:0]
    g0[3] = (unsigned)((ga >> 32) & 0x01ffffffu)       // global_addr[56:32]
          | (2u << 30);                                // type = 2 ("image")
    i32x8 g1;
    g1[0] = (int)(2u << 16);                           // data_size = 2 -> 4 bytes
    g1[1] = (int)(((unsigned)cols & 0xffffu) << 16);   // tensor_dim0[15:0]
    g1[2] = (int)(((unsigned)rows & 0xffffu) << 16);   // tensor_dim1[15:0] (dim0 hi = 0)
    g1[3] = (int)(((unsigned)cols & 0xffffu) << 16);   // tile_dim0 (dim1 hi = 0)
    g1[4] = (int)((unsigned)rows & 0xffffu);           // tile_dim1 (tile_dim2 = 0)
    g1[5] = ldb;                                       // tensor_dim0_stride[31:0]
    g1[6] = 0;                                         // stride hi / dim1_stride lo
    g1[7] = 0;
    i32x4 z4 = (i32x4)0;
    i32x8 z8 = (i32x8)0;
    __builtin_amdgcn_tensor_load_to_lds(g0, g1, z4, z4, z8, 0);
}

// ---------------- TDM-staged complex GEMM (B shared, n >= 64) --------------
// One workgroup = n/16 waves covering every 16-row stripe of one 32-column
// panel; the complex B panel is staged through LDS by the Tensor Data Mover
// in K-chunks of 64, then all waves consume it via ds loads.
template<bool HASAI, bool WANTI>
__global__ void __launch_bounds__(512) k_cgemm_tdmB(
    const float* __restrict__ Ar, const float* __restrict__ Ai, long long sA, int lda,
    const float* __restrict__ Br, const float* __restrict__ Bi, int ldb,
    float* __restrict__ Cr, float* __restrict__ Ci, long long sC, int ldc,
    int K, float alpha)
{
    const int KC = 64;
    __shared__ float shB[2 * KC * 32];                 // Re panel + Im panel (16 KB)
    float* shBr = shB;
    float* shBi = shB + KC * 32;

    const int lane = threadIdx.x & 31;
    const int wv   = threadIdx.x >> 5;
    const int n0   = blockIdx.x * 32;
    const int m0   = wv * 16;                          // waves cover the rows
    const long long z = blockIdx.z;

    const int r    = lane & 15;
    const int koff = (lane >> 4) << 1;

    v8f accR0 = (v8f)0.0f, accR1 = (v8f)0.0f;
    v8f accI0 = (v8f)0.0f, accI1 = (v8f)0.0f;

    const float* arow  = Ar + z * sA + (long long)(m0 + r) * lda + koff;
    const float* airow = HASAI ? (Ai + z * sA + (long long)(m0 + r) * lda + koff) : nullptr;

    const unsigned ldsR = (unsigned)(size_t)shBr;      // LDS aperture low bits = offset
    const unsigned ldsI = (unsigned)(size_t)shBi;

    for (int kc = 0; kc < K; kc += KC) {
        __syncthreads();                               // prev chunk fully consumed
        if (wv == 0) {
            tdm_load_panel(Br + (long long)kc * ldb + n0, ldsR, KC, 32, ldb);
            tdm_load_panel(Bi + (long long)kc * ldb + n0, ldsI, KC, 32, ldb);
            __builtin_amdgcn_s_wait_tensorcnt(0);
        }
        __syncthreads();                               // panel visible to all waves

        for (int k0 = 0; k0 < KC; k0 += 4) {
            v2f a;  a.x = arow[0]; a.y = arow[1];
            const int rB = (k0 + koff) * 32 + r;
            v2f b0; b0.x = shBr[rB];      b0.y = shBr[rB + 32];
            v2f b1; b1.x = shBr[rB + 16]; b1.y = shBr[rB + 48];
            v2f c0; c0.x = shBi[rB];      c0.y = shBi[rB + 32];
            v2f c1; c1.x = shBi[rB + 16]; c1.y = shBi[rB + 48];

            accR0 = __builtin_amdgcn_wmma_f32_16x16x4_f32(false, a, false, b0, (short)0, accR0, false, false);
            accR1 = __builtin_amdgcn_wmma_f32_16x16x4_f32(false, a, false, b1, (short)0, accR1, false, false);
            if (WANTI) {
                accI0 = __builtin_amdgcn_wmma_f32_16x16x4_f32(false, a, false, c0, (short)0, accI0, false, false);
                accI1 = __builtin_amdgcn_wmma_f32_16x16x4_f32(false, a, false, c1, (short)0, accI1, false, false);
            }
            if (HASAI) {
                v2f aI; aI.x = airow[0]; aI.y = airow[1];
                v2f aN; aN.x = -aI.x; aN.y = -aI.y;    // f32 WMMA: no A-neg modifier
                accR0 = __builtin_amdgcn_wmma_f32_16x16x4_f32(false, aN, false, c0, (short)0, accR0, false, false);
                accR1 = __builtin_amdgcn_wmma_f32_16x16x4_f32(false, aN, false, c1, (short)0, accR1, false, false);
                if (WANTI) {
                    accI0 = __builtin_amdgcn_wmma_f32_16x16x4_f32(false, aI, false, b0, (short)0, accI0, false, false);
                    accI1 = __builtin_amdgcn_wmma_f32_16x16x4_f32(false, aI, false, b1, (short)0, accI1, false, false);
                }
            }
            arow += 4;
            if (HASAI) airow += 4;
        }
    }

    const int rb = m0 + ((lane >> 4) << 3);
    float* cr = Cr + z * sC;
#pragma unroll
    for (int v = 0; v < 8; ++v) {
        cr[(long long)(rb + v) * ldc + n0 + r]      = alpha * accR0[v];
        cr[(long long)(rb + v) * ldc + n0 + 16 + r] = alpha * accR1[v];
    }
    if (WANTI) {
        float* ci = Ci + z * sC;
#pragma unroll
        for (int v = 0; v < 8; ++v) {
            ci[(long long)(rb + v) * ldc + n0 + r]      = alpha * accI0[v];
            ci[(long long)(rb + v) * ldc + n0 + 16 + r] = alpha * accI1[v];
        }
    }
}

// ------------------- batched complex GEMM via f32 WMMA ---------------------
// C = alpha * (A x B), square n x n matrices, row-major.
// One wave computes a 16 x (16*NT) tile; branch-free WMMA inner loop.
template<int NT, bool HASAI, bool HASBI, bool WANTI>
__global__ void __launch_bounds__(32) k_cgemm_t(
    const float* __restrict__ Ar, const float* __restrict__ Ai, long long sA, int lda,
    const float* __restrict__ Br, const float* __restrict__ Bi, long long sB, int ldb,
    float* __restrict__ Cr, float* __restrict__ Ci, long long sC, int ldc,
    int K, float alpha)
{
    const int lane = threadIdx.x;
    const int n0 = blockIdx.x * (16 * NT);
    const int m0 = blockIdx.y << 4;
    const long long z = blockIdx.z;

    const int r    = lane & 15;            // row of A frag / col of B,C frags
    const int koff = (lane >> 4) << 1;     // K sub-offset per half-wave

    v8f accR0 = (v8f)0.0f, accI0 = (v8f)0.0f;
    v8f accR1 = (v8f)0.0f, accI1 = (v8f)0.0f;

    const float* arow  = Ar + z * sA + (long long)(m0 + r) * lda + koff;
    const float* airow = HASAI ? (Ai + z * sA + (long long)(m0 + r) * lda + koff) : nullptr;
    const float* bp    = Br + z * sB + (long long)koff * ldb + n0 + r;
    const float* bpi   = HASBI ? (Bi + z * sB + (long long)koff * ldb + n0 + r) : nullptr;
    const long long bstep = 4ll * (long long)ldb;

    for (int k0 = 0; k0 < K; k0 += 4) {
        v2f a;  a.x  = arow[0]; a.y  = arow[1];
        v2f b0; b0.x = bp[0];   b0.y = bp[ldb];
        v2f b1;
        if (NT == 2) { b1.x = bp[16]; b1.y = bp[ldb + 16]; }

        accR0 = __builtin_amdgcn_wmma_f32_16x16x4_f32(false, a, false, b0, (short)0, accR0, false, false);
        if (NT == 2)
            accR1 = __builtin_amdgcn_wmma_f32_16x16x4_f32(false, a, false, b1, (short)0, accR1, false, false);

        if (HASBI) {
            v2f c0; c0.x = bpi[0]; c0.y = bpi[ldb];
            v2f c1;
            if (NT == 2) { c1.x = bpi[16]; c1.y = bpi[ldb + 16]; }
            if (WANTI) {
                accI0 = __builtin_amdgcn_wmma_f32_16x16x4_f32(false, a, false, c0, (short)0, accI0, false, false);
                if (NT == 2)
                    accI1 = __builtin_amdgcn_wmma_f32_16x16x4_f32(false, a, false, c1, (short)0, accI1, false, false);
            }
            if (HASAI) {
                v2f aI; aI.x = airow[0]; aI.y = airow[1];
                v2f aN; aN.x = -aI.x; aN.y = -aI.y;   // f32 WMMA: no A-neg modifier (ISA NEG = CNeg only)
                accR0 = __builtin_amdgcn_wmma_f32_16x16x4_f32(false, aN, false, c0, (short)0, accR0, false, false);
                if (NT == 2)
                    accR1 = __builtin_amdgcn_wmma_f32_16x16x4_f32(false, aN, false, c1, (short)0, accR1, false, false);
                if (WANTI) {
                    accI0 = __builtin_amdgcn_wmma_f32_16x16x4_f32(false, aI, false, b0, (short)0, accI0, false, false);
                    if (NT == 2)
                        accI1 = __builtin_amdgcn_wmma_f32_16x16x4_f32(false, aI, false, b1, (short)0, accI1, false, false);
                }
            }
        } else if (HASAI && WANTI) {
            v2f aI; aI.x = airow[0]; aI.y = airow[1];
            accI0 = __builtin_amdgcn_wmma_f32_16x16x4_f32(false, aI, false, b0, (short)0, accI0, false, false);
            if (NT == 2)
                accI1 = __builtin_amdgcn_wmma_f32_16x16x4_f32(false, aI, false, b1, (short)0, accI1, false, false);
        }

        arow += 4;
        if (HASAI) airow += 4;
        bp += bstep;
        if (HASBI) bpi += bstep;
    }

    const int rb = m0 + ((lane >> 4) << 3);
    float* cr = Cr + z * sC;
#pragma unroll
    for (int v = 0; v < 8; ++v) {
        cr[(long long)(rb + v) * ldc + n0 + r] = alpha * accR0[v];
        if (NT == 2) cr[(long long)(rb + v) * ldc + n0 + 16 + r] = alpha * accR1[v];
    }
    if (WANTI) {
        float* ci = Ci + z * sC;
#pragma unroll
        for (int v = 0; v < 8; ++v) {
            ci[(long long)(rb + v) * ldc + n0 + r] = alpha * accI0[v];
            if (NT == 2) ci[(long long)(rb + v) * ldc + n0 + 16 + r] = alpha * accI1[v];
        }
    }
}

// --------------------------- elementwise kernels ---------------------------
__global__ void k_mulf(const float* xr, const float* xi, long long sx, const float* f,
                       float* yr, float* yi, long long sy, int n2, int nb) {
    long long i = (long long)blockIdx.x * blockDim.x + threadIdx.x;
    if (i >= (long long)nb * n2) return;
    long long b = i / n2; int p = (int)(i % n2);
    float fv = f[p];
    yr[b * sy + p] = xr[b * sx + p] * fv;
    yi[b * sy + p] = xi[b * sx + p] * fv;
}

// band = j * x * f   ->  re' = -f*im, im' = f*re   (cfac = (-1j)^3 = +j)
__global__ void k_mulfj(const float* xr, const float* xi, long long sx, const float* f,
                        float* yr, float* yi, long long sy, int n2, int nb) {
    long long i = (long long)blockIdx.x * blockDim.x + threadIdx.x;
    if (i >= (long long)nb * n2) return;
    long long b = i / n2; int p = (int)(i % n2);
    float fv = f[p];
    yr[b * sy + p] = -xi[b * sx + p] * fv;
    yi[b * sy + p] =  xr[b * sx + p] * fv;
}

// center-crop of shifted spectrum == corner gather in unshifted layout
__global__ void k_crop(const float* xr, const float* xi, long long sx, int n,
                       const float* lo, float* yr, float* yi, long long sy, int nb) {
    int h = n >> 1;
    long long i = (long long)blockIdx.x * blockDim.x + threadIdx.x;
    if (i >= (long long)nb * h * h) return;
    long long b = i / ((long long)h * h); int p = (int)(i % ((long long)h * h));
    int yy = p / h, xx = p % h;
    int oy = (yy < (h >> 1)) ? yy : yy + h;
    int ox = (xx < (h >> 1)) ? xx : xx + h;
    float fv = lo[p];
    long long q = b * sx + (long long)oy * n + ox;
    yr[b * sy + p] = xr[q] * fv;
    yi[b * sy + p] = xi[q] * fv;
}

// spectral zero-pad upsampling (unshifted gather)
__global__ void k_upsamp(const float* xr, const float* xi, long long sx, int m,
                         float* yr, float* yi, long long sy, int nb) {
    int M = m << 1;
    long long i = (long long)blockIdx.x * blockDim.x + threadIdx.x;
    if (i >= (long long)nb * M * M) return;
    long long b = i / ((long long)M * M); int p = (int)(i % ((long long)M * M));
    int yy = p / M, xx = p % M;
    int syy = (yy < (m >> 1)) ? yy : ((yy >= m + (m >> 1)) ? yy - m : -1);
    int sxx = (xx < (m >> 1)) ? xx : ((xx >= m + (m >> 1)) ? xx - m : -1);
    float vr = 0.0f, vi = 0.0f;
    if (syy >= 0 && sxx >= 0) {
        long long q = b * sx + (long long)syy * m + sxx;
        vr = xr[q];
        vi = xi ? xi[q] : 0.0f;
    }
    yr[b * sy + p] = vr;
    yi[b * sy + p] = vi;
}

__global__ void k_cabs(const float* re, const float* im, float* mag, long long tot) {
    long long i = (long long)blockIdx.x * blockDim.x + threadIdx.x;
    if (i < tot) mag[i] = sqrtf(re[i] * re[i] + im[i] * im[i]);
}

__global__ void k_power(const float* re, const float* im, float* out, long long tot) {
    long long i = (long long)blockIdx.x * blockDim.x + threadIdx.x;
    if (i < tot) out[i] = re[i] * re[i] + im[i] * im[i];
}

__global__ void k_submean(float* x, int n2) {
    __shared__ float sh[256];
    float* v = x + (long long)blockIdx.x * n2;
    float s = 0.0f;
    for (int i = threadIdx.x; i < n2; i += 256) s += v[i];
    float mu = blockReduceSum(s, sh) / (float)n2;
    for (int i = threadIdx.x; i < n2; i += 256) v[i] -= mu;
}

__global__ void k_meansq(const float* x, int n2, float* out) {
    __shared__ float sh[256];
    const float* v = x + (long long)blockIdx.x * n2;
    float s = 0.0f;
    for (int i = threadIdx.x; i < n2; i += 256) { float a = v[i]; s += a * a; }
    float t = blockReduceSum(s, sh);
    if (threadIdx.x == 0) out[blockIdx.x] = t / (float)n2;
}

// mean / central moments 2,3,4 / min / max per plane
__global__ void k_moments(const float* x, long long pstride, int n2,
                          float* out, int outStride) {
    __shared__ float sh[256];
    const float* v = x + (long long)blockIdx.x * pstride;
    float s = 0.0f, mn = 3.4e38f, mx = -3.4e38f;
    for (int i = threadIdx.x; i < n2; i += 256) {
        float a = v[i]; s += a; mn = fminf(mn, a); mx = fmaxf(mx, a);
    }
    float tot = blockReduceSum(s, sh);
    sh[threadIdx.x] = mn; __syncthreads();
    for (int w = 128; w > 0; w >>= 1) { if (threadIdx.x < w) sh[threadIdx.x] = fminf(sh[threadIdx.x], sh[threadIdx.x + w]); __syncthreads(); }
    float mnall = sh[0]; __syncthreads();
    sh[threadIdx.x] = mx; __syncthreads();
    for (int w = 128; w > 0; w >>= 1) { if (threadIdx.x < w) sh[threadIdx.x] = fmaxf(sh[threadIdx.x], sh[threadIdx.x + w]); __syncthreads(); }
    float mxall = sh[0]; __syncthreads();
    float mu = tot / (float)n2;
    float c2 = 0.0f, c3 = 0.0f, c4 = 0.0f;
    for (int i = threadIdx.x; i < n2; i += 256) {
        float d = v[i] - mu; float d2 = d * d;
        c2 += d2; c3 += d2 * d; c4 += d2 * d2;
    }
    c2 = blockReduceSum(c2, sh);
    c3 = blockReduceSum(c3, sh);
    c4 = blockReduceSum(c4, sh);
    if (threadIdx.x == 0) {
        float* o = out + (long long)blockIdx.x * outStride;
        o[0] = mu; o[1] = c2 / n2; o[2] = c3 / n2; o[3] = c4 / n2; o[4] = mnall; o[5] = mxall;
    }
}

// 9x9 centered autocorrelation patch gather (unshifted wrap indices)
__global__ void k_acg(const float* ac, long long pstride, int n,
                      float* patch, int npl, int bStride,
                      float* var, int varStride) {
    int pl = blockIdx.x, t = threadIdx.x;
    const float* a = ac + (long long)pl * pstride;
    float v = a[0];
    int b = pl / npl, o = pl % npl;
    if (t < 81) {
        int i = t / 9, j = t % 9;
        int oy = (i - 4 + n) % n, ox = (j - 4 + n) % n;
        patch[(long long)b * bStride + o * 81 + t] = a[(long long)oy * n + ox] / v;
    }
    if (t == 0 && var) var[(long long)b * varStride] = v;
}

// cov[o][p] = sum(a_o * b_p)/n2, optionally normalized by sqrt(va*vb)
__global__ void k_xcorr(const float* A, const float* B, int nplA, int nplB, int n2,
                        const float* va, const float* vb,
                        float* out, int obStride, int base) {
    __shared__ float sh[256];
    int pair = blockIdx.x, b = blockIdx.y;
    int o = pair / nplB, p = pair % nplB;
    const float* ap = A + ((long long)b * nplA + o) * n2;
    const float* bp = B + ((long long)b * nplB + p) * n2;
    float s = 0.0f;
    for (int i = threadIdx.x; i < n2; i += 256) s += ap[i] * bp[i];
    float t = blockReduceSum(s, sh);
    if (threadIdx.x == 0) {
        float c = t / (float)n2;
        if (va) c /= sqrtf(va[b * nplA + o] * vb[b * nplB + p]);
        out[(long long)b * obStride + base + pair] = c;
    }
}

// doubled-phase planes: dbl = up^2/(|up|+eps), dmag = |up|
__global__ void k_dphase(const float* ur, const float* ui, int n2, int nb,
                         float* dbl, float* dmg) {
    long long tot = (long long)nb * 4 * n2;
    long long i = (long long)blockIdx.x * blockDim.x + threadIdx.x;
    if (i >= tot) return;
    long long pl = i / n2; int p = (int)(i % n2);
    long long b = pl >> 2; int o = (int)(pl & 3);
    float r = ur[pl * n2 + p], im = ui[pl * n2 + p];
    float m = sqrtf(r * r + im * im);
    float d = m + 1e-12f;
    dbl[((b * 8) + o) * (long long)n2 + p]     = (r * r - im * im) / d;
    dbl[((b * 8) + 4 + o) * (long long)n2 + p] = (2.0f * r * im) / d;
    dmg[pl * (long long)n2 + p] = m;
}

__global__ void k_reconsum(const float* up, const float* cre, float* rec, int n2, int nb) {
    long long i = (long long)blockIdx.x * blockDim.x + threadIdx.x;
    if (i >= (long long)nb * n2) return;
    long long b = i / n2; int p = (int)(i % n2);
    float s = up[b * (long long)n2 + p];
    for (int o = 0; o < 4; ++o) s += cre[((b * 4) + o) * (long long)n2 + p];
    rec[b * (long long)n2 + p] = s;
}

// ------------------------- final statistics vector -------------------------
// full[b][0..1930]: [pix(6), acm(1296), skew_r(5), kurt_r(5), acr(405),
//                    std_r(5), cocm(64), csm(48), csr(96), var_hp(1)]
__global__ void k_assemble(const float* pixm, const float* acm, const float* recm,
                           const float* vrec, const float* acr, const float* cocm,
                           const float* csm, const float* csr, const float* vhp,
                           float* full) {
    int f = blockIdx.x * blockDim.x + threadIdx.x;
    int b = blockIdx.y;
    if (f >= 1931) return;
    float val = 0.0f;
    if (f < 6) {
        const float* p = pixm + b * 6;
        if (f == 0) val = p[0];
        else if (f == 1) val = p[1];
        else if (f == 2) val = p[2] / powf(p[1], 1.5f);
        else if (f == 3) val = p[3] / (p[1] * p[1]);
        else if (f == 4) val = p[4];
        else val = p[5];
    } else if (f < 1302) {                 // acm (i,j,s,o)
        int t = f - 6;
        int o = t & 3, s = (t >> 2) & 3;
        int j = (t >> 4) % 9, i = t / 144;
        val = acm[(long long)b * 1296 + (s * 4 + o) * 81 + i * 9 + j];
    } else if (f < 1307) {                 // skew of recon
        int s = f - 1302;
        float v = vrec[b * 5 + s], vi = pixm[b * 6 + 1];
        float m3 = recm[b * 30 + s * 6 + 2];
        val = (v < 1e-6f * vi) ? 0.0f : m3 / powf(v, 1.5f);
    } else if (f < 1312) {                 // kurt of recon
        int s = f - 1307;
        float v = vrec[b * 5 + s], vi = pixm[b * 6 + 1];
        float m4 = recm[b * 30 + s * 6 + 3];
        val = (v < 1e-6f * vi) ? 3.0f : m4 / (v * v);
    } else if (f < 1717) {                 // acr (i,j,s)
        int t = f - 1312;
        int s = t % 5, r2 = t / 5;
        int j = r2 % 9, i = r2 / 9;
        val = acr[(long long)b * 405 + s * 81 + i * 9 + j];
    } else if (f < 1722) {                 // std of recon
        int s = f - 1717;
        val = sqrtf(vrec[b * 5 + s]);
    } else if (f < 1786) {                 // cocm (o,p,s)
        int t = f - 1722;
        int s = t & 3, p = (t >> 2) & 3, o = t >> 4;
        val = cocm[b * 64 + s * 16 + o * 4 + p];
    } else if (f < 1834) {                 // csm (o,p,s)
        int t = f - 1786;
        int s = t % 3, p = (t / 3) & 3, o = t / 12;
        val = csm[b * 48 + s * 16 + o * 4 + p];
    } else if (f < 1930) {                 // csr (o,p2,s)
        int t = f - 1834;
        int s = t % 3, p = (t / 3) & 7, o = t / 24;
        val = csr[b * 96 + s * 32 + o * 8 + p];
    } else {
        val = vhp[b];
    }
    full[(long long)b * 1931 + f] = val;
}

// compacted indices of non-redundant statistics (mirrors _mask_indices)
__global__ void k_mask(int* idx) {
    if (threadIdx.x | blockIdx.x) return;
    int c = 0;
    for (int f = 0; f < 1931; ++f) {
        bool keep = true;
        if (f >= 6 && f < 1302) {
            int t = f - 6, t3 = t / 16;
            int j = t3 % 9, i = t3 / 9;
            keep = (i >= j) && !((i == j) && (i >= 4));
        } else if (f >= 1312 && f < 1717) {
            int t = f - 1312, r2 = t / 5;
            int j = r2 % 9, i = r2 / 9;
            keep = (i >= j) && !((i == j) && (i >= 4));
        } else if (f >= 1722 && f < 1786) {
            int t = f - 1722;
            int p = (t >> 2) & 3, o = t >> 4;
            keep = (p <= o);
        }
        if (keep) idx[c++] = f;
    }
}

__global__ void k_gather(const float* full, const int* idx, float* out, int total, int K) {
    int i = blockIdx.x * blockDim.x + threadIdx.x;
    if (i >= total) return;
    int b = i / K, m = i % K;
    if (m >= 1931) { out[i] = 0.0f; return; }
    out[i] = full[(long long)b * 1931 + idx[m]];
}

// ---------------------------------------------------------------------------
extern "C" void kernel_launch(void* const* d_in, const int* in_sizes, int n_in,
                              void* d_out, int out_size, void* d_ws, size_t ws_size,
                              hipStream_t stream)
{
    (void)in_sizes; (void)n_in; (void)ws_size;
    const float* img = (const float*)d_in[0];    // (32,1,256,256) fp32
    float* out = (float*)d_out;

    static const int ns_[5] = {256, 128, 64, 32, 16};

    size_t off = 0;
    auto alloc = [&](size_t nfloats) -> float* {
        float* p = (float*)((char*)d_ws + off);
        off += ((nfloats * sizeof(float) + 255) & ~(size_t)255);
        return p;
    };

    // ---- constant tables ----
    float *Dr[5], *Di[5], *Ivr[5], *Ivi[5];
    for (int i = 0; i < 5; ++i) {
        size_t n2 = (size_t)ns_[i] * ns_[i];
        Dr[i] = alloc(n2); Di[i] = alloc(n2); Ivr[i] = alloc(n2); Ivi[i] = alloc(n2);
    }
    float* hi0 = alloc(65536);
    float* lo0 = alloc(65536);
    float* fb[4]; float* locrop[4];
    for (int s = 0; s < 4; ++s) {
        size_t n2 = (size_t)ns_[s] * ns_[s];
        fb[s] = alloc(4 * n2);
        int h = ns_[s] / 2;
        locrop[s] = alloc((size_t)h * h);
    }

    // ---- big buffers ----
    const size_t TP = (size_t)PSB * 4 * 65536;
    float* t0r = alloc(TP); float* t0i = alloc(TP);
    float* t1r = alloc(TP); float* t1i = alloc(TP);
    float* spr = alloc((size_t)PSB * 65536); float* spi = alloc((size_t)PSB * 65536);
    float* lAr = alloc((size_t)PSB * 65536); float* lAi = alloc((size_t)PSB * 65536);
    float* lBr = alloc((size_t)PSB * 65536); float* lBi = alloc((size_t)PSB * 65536);
    float *cfr[4], *cfi[4], *mg[4];
    for (int s = 0; s < 4; ++s) {
        size_t n2 = (size_t)ns_[s] * ns_[s];
        cfr[s] = alloc((size_t)PSB * 4 * n2);
        cfi[s] = alloc((size_t)PSB * 4 * n2);
        mg[s]  = alloc((size_t)PSB * 4 * n2);
    }
    float* rec[5];
    for (int s = 0; s < 5; ++s) rec[s] = alloc((size_t)PSB * ns_[s] * ns_[s]);
    float* hp  = alloc((size_t)PSB * 65536);
    float* upr = alloc((size_t)PSB * 4 * 65536);
    float* upi = alloc((size_t)PSB * 4 * 65536);
    float* dbl = alloc((size_t)PSB * 8 * 65536);
    float* dmg = alloc((size_t)PSB * 4 * 65536);

    // ---- stats buffers ----
    float* pixm = alloc(PSB * 6);
    float* acm  = alloc(PSB * 1296);
    float* acr  = alloc(PSB * 405);
    float* vrec = alloc(PSB * 5);
    float* recm = alloc(PSB * 30);
    float* cocm = alloc(PSB * 64);
    float* csm  = alloc(PSB * 48);
    float* csr  = alloc(PSB * 96);
    float* vmf  = alloc(PSB * 4); float* vmc = alloc(PSB * 4);
    float* vrf  = alloc(PSB * 4); float* vrc = alloc(PSB * 8);
    float* vhp  = alloc(PSB);
    float* full = alloc((size_t)PSB * 1931);
    int* midx   = (int*)alloc(2048);

    auto EW = [&](long long tot) { return dim3((unsigned)((tot + 255) / 256)); };

    // compile-time-specialized complex GEMM dispatch.
    // B shared across batch + complex + n>=64  ->  TDM/LDS-staged kernel.
    auto cgemm = [&](int n, int batch,
                     const float* Ar, const float* Ai, long long sA,
                     const float* Br, const float* Bi, long long sB,
                     float* Cr, float* Ci, long long sC, float alpha) {
        if (sB == 0 && Bi != nullptr && n >= 64) {
            dim3 g(n / 32, 1, batch), blk(32 * (n / 16), 1, 1);
            const int sel2 = (Ai ? 2 : 0) | (Ci ? 1 : 0);
            switch (sel2) {
                case 0: k_cgemm_tdmB<false, false><<<g, blk, 0, stream>>>(
                    Ar, Ai, sA, n, Br, Bi, n, Cr, Ci, sC, n, n, alpha); break;
                case 1: k_cgemm_tdmB<false, true><<<g, blk, 0, stream>>>(
                    Ar, Ai, sA, n, Br, Bi, n, Cr, Ci, sC, n, n, alpha); break;
                case 2: k_cgemm_tdmB<true, false><<<g, blk, 0, stream>>>(
                    Ar, Ai, sA, n, Br, Bi, n, Cr, Ci, sC, n, n, alpha); break;
                default: k_cgemm_tdmB<true, true><<<g, blk, 0, stream>>>(
                    Ar, Ai, sA, n, Br, Bi, n, Cr, Ci, sC, n, n, alpha); break;
            }
            return;
        }
        const int nt = (n % 32 == 0) ? 2 : 1;
        dim3 g(n / (16 * nt), n / 16, batch), blk(32, 1, 1);
        const int sel = (Ai ? 4 : 0) | (Bi ? 2 : 0) | (Ci ? 1 : 0);
#define CG_CASE(NTV, SEL, A, B, C) \
        case SEL: k_cgemm_t<NTV, A, B, C><<<g, blk, 0, stream>>>( \
            Ar, Ai, sA, n, Br, Bi, sB, n, Cr, Ci, sC, n, n, alpha); break;
        if (nt == 2) {
            switch (sel) {
                CG_CASE(2, 0, false, false, false)
                CG_CASE(2, 1, false, false, true)
                CG_CASE(2, 2, false, true,  false)
                CG_CASE(2, 3, false, true,  true)
                CG_CASE(2, 4, true,  false, false)
                CG_CASE(2, 5, true,  false, true)
                CG_CASE(2, 6, true,  true,  false)
                CG_CASE(2, 7, true,  true,  true)
            }
        } else {
            switch (sel) {
                CG_CASE(1, 0, false, false, false)
                CG_CASE(1, 1, false, false, true)
                CG_CASE(1, 2, false, true,  false)
                CG_CASE(1, 3, false, true,  true)
                CG_CASE(1, 4, true,  false, false)
                CG_CASE(1, 5, true,  false, true)
                CG_CASE(1, 6, true,  true,  false)
                CG_CASE(1, 7, true,  true,  true)
            }
        }
#undef CG_CASE
    };

    // ---- constants ----
    for (int i = 0; i < 5; ++i) {
        int n = ns_[i], n2 = n * n;
        k_make_dft<<<(n2 + 255) / 256, 256, 0, stream>>>(Dr[i], Di[i], Ivr[i], Ivi[i], n);
    }
    k_radial<<<(65536 + 255) / 256, 256, 0, stream>>>(hi0, 256, 1.0f, 1);
    k_radial<<<(65536 + 255) / 256, 256, 0, stream>>>(lo0, 256, 1.0f, 0);
    for (int s = 0; s < 4; ++s) {
        int n = ns_[s], n2 = n * n;
        k_bands<<<(n2 + 255) / 256, 256, 0, stream>>>(fb[s], n);
        int h = n / 2;
        k_radial<<<(h * h + 255) / 256, 256, 0, stream>>>(locrop[s], h, 1.0f, 0);
    }

    // ---- forward FFT of images (fft2 = D*(X*D), D symmetric) ----
    cgemm(256, PSB, img, nullptr, 65536, Dr[0], Di[0], 0, t0r, t0i, 65536, 1.0f);
    cgemm(256, PSB, Dr[0], Di[0], 0, t0r, t0i, 65536, spr, spi, 65536, 1.0f);

    // ---- highpass = Re(ifft2(dft*hi0)) ----
    k_mulf<<<EW((long long)PSB * 65536), 256, 0, stream>>>(spr, spi, 65536, hi0, t1r, t1i, 65536, 65536, PSB);
    cgemm(256, PSB, t1r, t1i, 65536, Ivr[0], Ivi[0], 0, t0r, t0i, 65536, 1.0f);
    cgemm(256, PSB, Ivr[0], Ivi[0], 0, t0r, t0i, 65536, hp, nullptr, 65536, 1.0f);

    // ---- steerable pyramid ----
    k_mulf<<<EW((long long)PSB * 65536), 256, 0, stream>>>(spr, spi, 65536, lo0, lAr, lAi, 65536, 65536, PSB);
    float *cLr = lAr, *cLi = lAi, *nLr = lBr, *nLi = lBi;
    for (int s = 0; s < 4; ++s) {
        int n = ns_[s]; long long n2 = (long long)n * n;
        for (int o = 0; o < 4; ++o) {
            k_mulfj<<<EW((long long)PSB * n2), 256, 0, stream>>>(cLr, cLi, 65536, fb[s] + o * n2, t1r, t1i, n2, (int)n2, PSB);
            cgemm(n, PSB, t1r, t1i, n2, Ivr[s], Ivi[s], 0, t0r, t0i, n2, 1.0f);
            cgemm(n, PSB, Ivr[s], Ivi[s], 0, t0r, t0i, n2, cfr[s] + o * n2, cfi[s] + o * n2, 4 * n2, 1.0f);
        }
        int h = n / 2;
        k_crop<<<EW((long long)PSB * h * h), 256, 0, stream>>>(cLr, cLi, 65536, n, locrop[s], nLr, nLi, 65536, PSB);
        { float* t = cLr; cLr = nLr; nLr = t; t = cLi; cLi = nLi; nLi = t; }
    }
    // lowpass = recon[4]
    cgemm(16, PSB, cLr, cLi, 65536, Ivr[4], Ivi[4], 0, t0r, t0i, 256, 1.0f);
    cgemm(16, PSB, Ivr[4], Ivi[4], 0, t0r, t0i, 256, rec[4], nullptr, 256, 1.0f);

    // ---- magnitudes (mean-subtracted) ----
    for (int s = 0; s < 4; ++s) {
        long long n2 = (long long)ns_[s] * ns_[s];
        k_cabs<<<EW((long long)PSB * 4 * n2), 256, 0, stream>>>(cfr[s], cfi[s], mg[s], (long long)PSB * 4 * n2);
        k_submean<<<PSB * 4, 256, 0, stream>>>(mg[s], (int)n2);
    }

    // ---- reconstruction chain (spectral upsampling) ----
    for (int s = 3; s >= 0; --s) {
        int m = ns_[s + 1], M = ns_[s];
        long long m2 = (long long)m * m, M2 = (long long)M * M;
        cgemm(m, PSB, rec[s + 1], nullptr, m2, Dr[s + 1], Di[s + 1], 0, t0r, t0i, m2, 1.0f);
        cgemm(m, PSB, Dr[s + 1], Di[s + 1], 0, t0r, t0i, m2, t1r, t1i, m2, 1.0f);
        k_upsamp<<<EW((long long)PSB * M2), 256, 0, stream>>>(t1r, t1i, m2, m, t0r, t0i, M2, PSB);
        cgemm(M, PSB, t0r, t0i, M2, Ivr[s], Ivi[s], 0, t1r, t1i, M2, 1.0f);
        cgemm(M, PSB, Ivr[s], Ivi[s], 0, t1r, t1i, M2, t0r, nullptr, M2, 4.0f);
        k_reconsum<<<EW((long long)PSB * M2), 256, 0, stream>>>(t0r, cfr[s], rec[s], (int)M2, PSB);
    }

    // ---- recon moments + FFT autocorrelation ----
    for (int s = 0; s < 5; ++s) {
        int n = ns_[s]; long long n2 = (long long)n * n;
        k_moments<<<PSB, 256, 0, stream>>>(rec[s], n2, (int)n2, recm + s * 6, 30);
        cgemm(n, PSB, rec[s], nullptr, n2, Dr[s], Di[s], 0, t0r, t0i, n2, 1.0f);
        cgemm(n, PSB, Dr[s], Di[s], 0, t0r, t0i, n2, t1r, t1i, n2, 1.0f);
        k_power<<<EW((long long)PSB * n2), 256, 0, stream>>>(t1r, t1i, t0r, (long long)PSB * n2);
        cgemm(n, PSB, t0r, nullptr, n2, Ivr[s], Ivi[s], 0, t1r, t1i, n2, 1.0f);
        cgemm(n, PSB, Ivr[s], Ivi[s], 0, t1r, t1i, n2, t0r, nullptr, n2, 1.0f / (float)n2);
        k_acg<<<PSB, 96, 0, stream>>>(t0r, n2, n, acr + s * 81, 1, 405, vrec + s, 5);
    }

    // ---- magnitude autocorrelation + same-scale cross-correlation ----
    for (int s = 0; s < 4; ++s) {
        int n = ns_[s]; long long n2 = (long long)n * n; int batch = PSB * 4;
        cgemm(n, batch, mg[s], nullptr, n2, Dr[s], Di[s], 0, t0r, t0i, n2, 1.0f);
        cgemm(n, batch, Dr[s], Di[s], 0, t0r, t0i, n2, t1r, t1i, n2, 1.0f);
        k_power<<<EW((long long)batch * n2), 256, 0, stream>>>(t1r, t1i, t0r, (long long)batch * n2);
        cgemm(n, batch, t0r, nullptr, n2, Ivr[s], Ivi[s], 0, t1r, t1i, n2, 1.0f);
        cgemm(n, batch, Ivr[s], Ivi[s], 0, t1r, t1i, n2, t0r, nullptr, n2, 1.0f / (float)n2);
        k_acg<<<batch, 96, 0, stream>>>(t0r, n2, n, acm + s * 324, 4, 1296, nullptr, 0);
        k_xcorr<<<dim3(16, PSB), 256, 0, stream>>>(mg[s], mg[s], 4, 4, (int)n2, nullptr, nullptr, cocm, 64, s * 16);
    }

    // ---- cross-scale statistics (doubled-phase) ----
    for (int s = 0; s < 3; ++s) {
        int m = ns_[s + 1], M = ns_[s];
        long long m2 = (long long)m * m, M2 = (long long)M * M;
        int batch = PSB * 4;
        cgemm(m, batch, cfr[s + 1], cfi[s + 1], m2, Dr[s + 1], Di[s + 1], 0, t0r, t0i, m2, 1.0f);
        cgemm(m, batch, Dr[s + 1], Di[s + 1], 0, t0r, t0i, m2, t1r, t1i, m2, 1.0f);
        k_upsamp<<<EW((long long)batch * M2), 256, 0, stream>>>(t1r, t1i, m2, m, t0r, t0i, M2, batch);
        cgemm(M, batch, t0r, t0i, M2, Ivr[s], Ivi[s], 0, t1r, t1i, M2, 1.0f);
        cgemm(M, batch, Ivr[s], Ivi[s], 0, t1r, t1i, M2, upr, upi, M2, 4.0f);
        k_dphase<<<EW((long long)PSB * 4 * M2), 256, 0, stream>>>(upr, upi, (int)M2, PSB, dbl, dmg);
        k_submean<<<PSB * 4, 256, 0, stream>>>(dmg, (int)M2);
        k_meansq<<<PSB * 4, 256, 0, stream>>>(mg[s], (int)M2, vmf);
        k_meansq<<<PSB * 4, 256, 0, stream>>>(dmg, (int)M2, vmc);
        k_meansq<<<PSB * 4, 256, 0, stream>>>(cfr[s], (int)M2, vrf);
        k_meansq<<<PSB * 8, 256, 0, stream>>>(dbl, (int)M2, vrc);
        k_xcorr<<<dim3(16, PSB), 256, 0, stream>>>(mg[s], dmg, 4, 4, (int)M2, vmf, vmc, csm, 48, s * 16);
        k_xcorr<<<dim3(32, PSB), 256, 0, stream>>>(cfr[s], dbl, 4, 8, (int)M2, vrf, vrc, csr, 96, s * 32);
    }

    // ---- highpass variance, pixel stats, assembly ----
    k_meansq<<<PSB, 256, 0, stream>>>(hp, 65536, vhp);
    k_moments<<<PSB, 256, 0, stream>>>(img, 65536, 65536, pixm, 6);

    k_assemble<<<dim3(8, PSB), 256, 0, stream>>>(pixm, acm, recm, vrec, acr, cocm, csm, csr, vhp, full);
    k_mask<<<1, 1, 0, stream>>>(midx);
    int K = out_size / PSB;
    if (K > 0)
        k_gather<<<(out_size + 255) / 256, 256, 0, stream>>>(full, midx, out, out_size, K);
}